// GHMC_11596411700015
// MI455X (gfx1250) — compile-verified
//
#include <hip/hip_runtime.h>

// GHMC loss for MI455X (gfx1250, wave32).
// Memory-bound: ~100 MB/pass, 2 passes; pass 2 is L2-resident (192 MB L2).
// No matmul in the reference -> no WMMA; optimize the streaming + histogram path.

#define BINS   10
#define TPB    256          // 8 wave32 per block
#define NBLK   2048         // 2048 * 256 = 524288 threads; 16 samples/thread at N=8.4M
#define WAVES  (TPB / 32)

__device__ __forceinline__ float fast_sigmoid(float x) {
    return 1.0f / (1.0f + __expf(-x));
}

// ---------------------------------------------------------------- zero counts
__global__ void ghmc_zero(unsigned* __restrict__ counts) {
    if (threadIdx.x < BINS) counts[threadIdx.x] = 0u;
}

// ---------------------------------------------------------------- pass 1: histogram
// Per-wave LDS sub-histograms (padded to 16 words so waves never share a bank row),
// then one global atomic per (block, bin).
__global__ void ghmc_hist(const float* __restrict__ pred,
                          const int*   __restrict__ tgt,
                          unsigned*    __restrict__ gcounts,
                          int nsamp) {
    __shared__ unsigned hist[WAVES * 16];
    const int tid  = threadIdx.x;
    const int wave = tid >> 5;
    if (tid < WAVES * 16) hist[tid] = 0u;
    __syncthreads();

    unsigned* h = &hist[wave * 16];
    const int stride = gridDim.x * blockDim.x;
    const int npairs = nsamp >> 1;              // 2 samples / iteration
    const float4* __restrict__ p4 = (const float4*)pred;
    const int2*   __restrict__ t2 = (const int2*)tgt;

    for (int i = blockIdx.x * blockDim.x + tid; i < npairs; i += stride) {
        __builtin_prefetch(&p4[i + stride], 0, 0);   // global_prefetch_b8
        float4 p = p4[i];                             // global_load_b128
        int2   t = t2[i];                             // global_load_b64

        // sample A: logits (p.x, p.y), label t.x
        float gA0 = fast_sigmoid(t.x == 0 ? -p.x : p.x);   // |sig - onehot0|
        float gA1 = fast_sigmoid(t.x == 1 ? -p.y : p.y);   // |sig - onehot1|
        // sample B: logits (p.z, p.w), label t.y
        float gB0 = fast_sigmoid(t.y == 0 ? -p.z : p.z);
        float gB1 = fast_sigmoid(t.y == 1 ? -p.w : p.w);

        int bA0 = min((int)(gA0 * (float)BINS), BINS - 1);
        int bA1 = min((int)(gA1 * (float)BINS), BINS - 1);
        int bB0 = min((int)(gB0 * (float)BINS), BINS - 1);
        int bB1 = min((int)(gB1 * (float)BINS), BINS - 1);

        atomicAdd(&h[bA0], 1u);   // ds_add_u32
        atomicAdd(&h[bA1], 1u);
        atomicAdd(&h[bB0], 1u);
        atomicAdd(&h[bB1], 1u);
    }
    __syncthreads();

    if (tid < BINS) {
        unsigned s = 0;
        #pragma unroll
        for (int w = 0; w < WAVES; ++w) s += hist[w * 16 + tid];
        atomicAdd(&gcounts[tid], s);              // global_atomic_add_u32
    }
}

// ---------------------------------------------------------------- weight table
// Combined per-element multiplier:
//   (tot / counts[b]) / n_nonempty / tot  ==  1 / (counts[b] * n_nonempty)
__global__ void ghmc_weights(const unsigned* __restrict__ gcounts,
                             float* __restrict__ wtab) {
    if (threadIdx.x == 0 && blockIdx.x == 0) {
        int ne = 0;
        #pragma unroll
        for (int b = 0; b < BINS; ++b) ne += (gcounts[b] > 0u) ? 1 : 0;
        float inv_ne = (ne > 0) ? (1.0f / (float)ne) : 1.0f;
        #pragma unroll
        for (int b = 0; b < BINS; ++b) {
            unsigned c = gcounts[b];
            wtab[b] = (c > 0u) ? (inv_ne / (float)c) : 0.0f;
        }
    }
}

// ---------------------------------------------------------------- pass 2: weighted BCE sum
__device__ __forceinline__ float sample_loss(float x0, float x1, int t,
                                             const float* __restrict__ wlds) {
    float g0 = fast_sigmoid(t == 0 ? -x0 : x0);
    float g1 = fast_sigmoid(t == 1 ? -x1 : x1);
    int b0 = min((int)(g0 * (float)BINS), BINS - 1);
    int b1 = min((int)(g1 * (float)BINS), BINS - 1);
    float t0 = (t == 0) ? 1.0f : 0.0f;
    float t1 = (t == 1) ? 1.0f : 0.0f;
    // stable BCE-with-logits: max(x,0) - x*t + log1p(exp(-|x|))
    float bce0 = fmaxf(x0, 0.0f) - x0 * t0 + log1pf(__expf(-fabsf(x0)));
    float bce1 = fmaxf(x1, 0.0f) - x1 * t1 + log1pf(__expf(-fabsf(x1)));
    return wlds[b0] * bce0 + wlds[b1] * bce1;   // ds_load_b32 lookups
}

__global__ void ghmc_loss(const float* __restrict__ pred,
                          const int*   __restrict__ tgt,
                          const float* __restrict__ wtab,
                          float*       __restrict__ partials,
                          int nsamp) {
    __shared__ float wlds[16];
    __shared__ float red[WAVES];
    const int tid = threadIdx.x;
    if (tid < BINS) wlds[tid] = wtab[tid];
    else if (tid < 16) wlds[tid] = 0.0f;
    __syncthreads();

    const int stride = gridDim.x * blockDim.x;
    const int npairs = nsamp >> 1;
    const float4* __restrict__ p4 = (const float4*)pred;
    const int2*   __restrict__ t2 = (const int2*)tgt;

    float acc = 0.0f;
    for (int i = blockIdx.x * blockDim.x + tid; i < npairs; i += stride) {
        __builtin_prefetch(&p4[i + stride], 0, 0);
        float4 p = p4[i];
        int2   t = t2[i];
        acc += sample_loss(p.x, p.y, t.x, wlds);
        acc += sample_loss(p.z, p.w, t.y, wlds);
    }

    // wave32 butterfly reduce (deterministic), then cross-wave via LDS
    #pragma unroll
    for (int off = 16; off > 0; off >>= 1) acc += __shfl_xor(acc, off, 32);
    if ((tid & 31) == 0) red[tid >> 5] = acc;
    __syncthreads();
    if (tid == 0) {
        float s = 0.0f;
        #pragma unroll
        for (int w = 0; w < WAVES; ++w) s += red[w];
        partials[blockIdx.x] = s;
    }
}

// ---------------------------------------------------------------- final reduce
__global__ void ghmc_final(const float* __restrict__ partials,
                           float* __restrict__ out) {
    __shared__ float red[WAVES];
    const int tid = threadIdx.x;
    float acc = 0.0f;
    for (int i = tid; i < NBLK; i += TPB) acc += partials[i];
    #pragma unroll
    for (int off = 16; off > 0; off >>= 1) acc += __shfl_xor(acc, off, 32);
    if ((tid & 31) == 0) red[tid >> 5] = acc;
    __syncthreads();
    if (tid == 0) {
        float s = 0.0f;
        #pragma unroll
        for (int w = 0; w < WAVES; ++w) s += red[w];
        out[0] = s;   // LOSS_WEIGHT == 1.0; all other scaling folded into wtab
    }
}

// ---------------------------------------------------------------- launch
extern "C" void kernel_launch(void* const* d_in, const int* in_sizes, int n_in,
                              void* d_out, int out_size, void* d_ws, size_t ws_size,
                              hipStream_t stream) {
    const float* pred = (const float*)d_in[0];   // [N, 2] float32
    const int*   tgt  = (const int*)d_in[1];     // [N]    int
    const int nsamp = in_sizes[1];               // N samples (pred has 2N floats)

    // workspace layout (ws_size >= (32 + NBLK) * 4 bytes = 8.3 KB)
    unsigned* counts   = (unsigned*)d_ws;            // [0..15]  bin counts
    float*    wtab     = (float*)d_ws + 16;          // [16..31] weight table
    float*    partials = (float*)d_ws + 32;          // [32..32+NBLK)

    ghmc_zero   <<<1,    32,  0, stream>>>(counts);
    ghmc_hist   <<<NBLK, TPB, 0, stream>>>(pred, tgt, counts, nsamp);
    ghmc_weights<<<1,    32,  0, stream>>>(counts, wtab);
    ghmc_loss   <<<NBLK, TPB, 0, stream>>>(pred, tgt, wtab, partials, nsamp);
    ghmc_final  <<<1,    TPB, 0, stream>>>(partials, (float*)d_out);
}